// MaximalCodingRateReduction_41704132444731
// MI455X (gfx1250) — compile-verified
//
#include <hip/hip_runtime.h>
#include <hip/hip_bf16.h>

// MCR^2 loss for MI455X (gfx1250).
// Heavy op: class-partitioned SYRK  cov[k] = X^T diag(1[y=k]) X  done with
// V_WMMA_F32_16X16X4_F32 (fp32 matrix cores, wave32). 11 Cholesky log-dets
// done with the full 256x256 matrix resident in LDS (320KB/WGP on CDNA5).

typedef __attribute__((ext_vector_type(2))) float v2f;
typedef __attribute__((ext_vector_type(8))) float v8f;

#define NSAMP 65536
#define DDIM  256
#define KCLS  10
#define EPSV  0.01f

// ---------------------------------------------------------------- count
__global__ __launch_bounds__(256) void mcr2_count_kernel(const int* __restrict__ Y,
                                                         int* __restrict__ counts,
                                                         float* __restrict__ trPi) {
    __shared__ int sc[KCLS];
    if (threadIdx.x < KCLS) sc[threadIdx.x] = 0;
    __syncthreads();
    for (int idx = threadIdx.x; idx < NSAMP; idx += blockDim.x) {
        int y = Y[idx];
        if (y >= 0 && y < KCLS) atomicAdd(&sc[y], 1);
    }
    __syncthreads();
    if (threadIdx.x < KCLS) {
        counts[threadIdx.x] = sc[threadIdx.x];
        trPi[threadIdx.x]   = (float)sc[threadIdx.x] + 1e-8f;
    }
}

// ---------------------------------------------------------------- SYRK (WMMA)
// grid.x = 136 triangular tiles (i<=j), grid.y = S sample slices, block = 1 wave.
// Ten v8f accumulators (one per class) stay pinned in VGPRs; per 4-sample chunk
// a scalar presence bitmask selects which class accumulators receive a masked
// rank-4 WMMA update. Exactly one WMMA site per accumulator -> no phi copies.
__global__ __launch_bounds__(32) void mcr2_syrk_kernel(const float* __restrict__ X,
                                                       const int* __restrict__ Y,
                                                       float* __restrict__ part,
                                                       int sliceLen) {
    // decode triangular tile id -> (i,j) with i<=j
    int t = blockIdx.x;
    int i = 0;
    while (t >= (16 - i)) { t -= (16 - i); ++i; }
    int j = i + t;

    const int lane = threadIdx.x;        // 0..31
    const int half = lane >> 4;          // 0: samples {0,1}; 1: samples {2,3}
    const int l15  = lane & 15;
    const int colI = i * 16 + l15;
    const int colJ = j * 16 + l15;

    v8f acc[KCLS];
#pragma unroll
    for (int k = 0; k < KCLS; ++k)
#pragma unroll
        for (int r = 0; r < 8; ++r) acc[k][r] = 0.0f;

    const int base0 = blockIdx.y * sliceLen;

    for (int c = 0; c < sliceLen; c += 4) {
        const int base = base0 + c;
        // 4 labels, made wave-uniform (SGPR) so the per-class guards are scalar
        // branches and EXEC stays all-1s for WMMA.
        const int4 yv = *(const int4*)(Y + base);
        const int y0 = __builtin_amdgcn_readfirstlane(yv.x);
        const int y1 = __builtin_amdgcn_readfirstlane(yv.y);
        const int y2 = __builtin_amdgcn_readfirstlane(yv.z);
        const int y3 = __builtin_amdgcn_readfirstlane(yv.w);
        const unsigned pres = (1u << y0) | (1u << y1) | (1u << y2) | (1u << y3);

        const int la = half ? y2 : y0;   // label of this lane's first sample
        const int lb = half ? y3 : y1;   // label of this lane's second sample

        const float* r0 = X + (size_t)(base + (half ? 2 : 0)) * DDIM;
        const float* r1 = r0 + DDIM;
        const float ax = r0[colI], ay = r1[colI];
        const float bx = r0[colJ], by = r1[colJ];

#pragma unroll
        for (int k = 0; k < KCLS; ++k) {
            if (pres & (1u << k)) {              // scalar test on SGPR bitmask
                v2f am, bm;
                am.x = (la == k) ? ax : 0.0f;
                am.y = (lb == k) ? ay : 0.0f;
                bm.x = (la == k) ? bx : 0.0f;
                bm.y = (lb == k) ? by : 0.0f;
                acc[k] = __builtin_amdgcn_wmma_f32_16x16x4_f32(
                    false, am, false, bm, (short)0, acc[k], false, false);
            }
        }
    }

    // write partial tiles: part[slice][class][256*256], only i<=j tiles populated
    const size_t pbase = (size_t)blockIdx.y * KCLS * (DDIM * DDIM);
#pragma unroll
    for (int k = 0; k < KCLS; ++k) {
#pragma unroll
        for (int r = 0; r < 8; ++r) {
            const int row = i * 16 + r + (half ? 8 : 0);
            const int col = j * 16 + l15;
            part[pbase + (size_t)k * (DDIM * DDIM) + row * DDIM + col] = acc[k][r];
        }
    }
}

// ---------------------------------------------------------------- reduce + mirror
__global__ __launch_bounds__(256) void mcr2_reduce_kernel(const float* __restrict__ part,
                                                          float* __restrict__ cov,
                                                          int S) {
    int idx = blockIdx.x * blockDim.x + threadIdx.x;   // over KCLS * 256 * 256
    if (idx >= KCLS * DDIM * DDIM) return;
    const int k  = idx >> 16;
    const int rc = idx & 65535;
    const int r = rc >> 8, c = rc & 255;
    const int ti = r >> 4, tj = c >> 4;
    const int src = (ti <= tj) ? (r * DDIM + c) : (c * DDIM + r);
    float s = 0.0f;
    for (int w = 0; w < S; ++w)
        s += part[((size_t)w * KCLS + k) * (DDIM * DDIM) + src];
    cov[idx] = s;
}

// ---------------------------------------------------------------- Cholesky logdet
// blocks 0..9: I + s_k*cov[k]; block 10: I + scalar*gram (gram = sum_k cov[k])
__global__ __launch_bounds__(256) void mcr2_logdet_kernel(const float* __restrict__ cov,
                                                          const float* __restrict__ trPi,
                                                          float* __restrict__ logdets) {
    extern __shared__ float sA[];        // 256 x 257 (padded: conflict-free columns)
    __shared__ float sDiag;
    const int b   = blockIdx.x;
    const int tid = threadIdx.x;

    float s;
    if (b < KCLS) s = (float)DDIM / (trPi[b] * EPSV);
    else          s = (float)DDIM / ((float)NSAMP * EPSV);   // GAM1 == 1

    for (int idx = tid; idx < DDIM * DDIM; idx += 256) {
        const int r = idx >> 8, c = idx & 255;
        float v;
        if (b < KCLS) {
            v = cov[(size_t)b * (DDIM * DDIM) + idx];
        } else {
            v = 0.0f;
#pragma unroll
            for (int k = 0; k < KCLS; ++k) v += cov[(size_t)k * (DDIM * DDIM) + idx];
        }
        sA[r * 257 + c] = s * v + ((r == c) ? 1.0f : 0.0f);
    }

    float logacc = 0.0f;                 // meaningful in thread 0 only
    for (int j = 0; j < DDIM; ++j) {
        __syncthreads();
        if (tid == 0) {
            const float d = sqrtf(sA[j * 257 + j]);
            sDiag = d;
            logacc += logf(d);
        }
        __syncthreads();
        const float d = sDiag;
        if (tid > j) sA[tid * 257 + j] /= d;     // scaled column L[:,j]
        if (tid == j) sA[j * 257 + j] = d;
        __syncthreads();
        if (tid > j) {
            const float lij = sA[tid * 257 + j];
            for (int k = j + 1; k < DDIM; ++k)
                sA[tid * 257 + k] -= lij * sA[k * 257 + j];   // broadcast column read
        }
    }
    __syncthreads();
    if (tid == 0) logdets[b] = 2.0f * logacc;
}

// ---------------------------------------------------------------- final scalars
__global__ void mcr2_final_kernel(const float* __restrict__ logdets,
                                  const float* __restrict__ trPi,
                                  float* __restrict__ out) {
    if (threadIdx.x != 0 || blockIdx.x != 0) return;
    const float disc = 0.5f * logdets[KCLS];     // gram logdet
    float comp = 0.0f;
    for (int k = 0; k < KCLS; ++k) comp += logdets[k] * trPi[k];
    comp /= (2.0f * (float)NSAMP);
    out[0] = -disc + comp;   // GAM2 = 1
    out[1] = disc;           // discrimn_empi
    out[2] = comp;           // compress_empi
    out[3] = disc;           // discrimn_theo (GAM1 = 1 -> identical)
    out[4] = comp;           // compress_theo (identical)
}

// ---------------------------------------------------------------- host launcher
extern "C" void kernel_launch(void* const* d_in, const int* in_sizes, int n_in,
                              void* d_out, int out_size, void* d_ws, size_t ws_size,
                              hipStream_t stream) {
    (void)in_sizes; (void)n_in; (void)out_size;
    const float* X = (const float*)d_in[0];
    const int*   Y = (const int*)d_in[1];
    float* out = (float*)d_out;

    char* ws = (char*)d_ws;
    int*   counts  = (int*)ws;                 // 16 ints
    float* trPi    = (float*)(ws + 64);        // 16 floats
    float* logdets = (float*)(ws + 128);       // 16 floats
    float* cov     = (float*)(ws + 256);       // 10 * 256*256 floats
    const size_t covB = (size_t)KCLS * DDIM * DDIM * sizeof(float);
    float* part    = (float*)(ws + 256 + covB);

    // pick split-K slice count that fits the workspace (deterministic per run)
    int S = 16;
    while (S > 1 && (256 + covB + (size_t)S * covB) > ws_size) S >>= 1;
    const int sliceLen = NSAMP / S;

    mcr2_count_kernel<<<1, 256, 0, stream>>>(Y, counts, trPi);

    dim3 sg(136, (unsigned)S);
    mcr2_syrk_kernel<<<sg, 32, 0, stream>>>(X, Y, part, sliceLen);

    const int nred = KCLS * DDIM * DDIM;
    mcr2_reduce_kernel<<<(nred + 255) / 256, 256, 0, stream>>>(part, cov, S);

    mcr2_logdet_kernel<<<KCLS + 1, 256, 257 * DDIM * sizeof(float), stream>>>(cov, trPi, logdets);

    mcr2_final_kernel<<<1, 1, 0, stream>>>(logdets, trPi, out);
}